// BoundaryLoss_36103495090313
// MI455X (gfx1250) — compile-verified
//
#include <hip/hip_runtime.h>
#include <math.h>

// Problem constants (reference: pred/target [16,1,256,256] f32, scalar f32 out)
#define HH 256
#define WW 256
#define BATCH 16
#define IMG (HH * WW)        // 65536
#define NTOT (BATCH * IMG)   // 1048576
#define INF_D 1.0e6f

typedef __attribute__((ext_vector_type(2))) float v2f;
typedef __attribute__((ext_vector_type(8))) float v8f;

__device__ __forceinline__ float fast_sigmoid(float x) {
  // v_exp_f32 + v_rcp_f32 (native), avoids IEEE div expansion in the hot loop
  return __builtin_amdgcn_rcpf(1.0f + __expf(-x));
}

// ---------------------------------------------------------------------------
// Kernel 0: zero the atomic-max slots and the output accumulator
// ---------------------------------------------------------------------------
__global__ void bl_init(unsigned* __restrict__ maxabs, float* __restrict__ out) {
  int t = threadIdx.x;
  if (t < BATCH) maxabs[t] = 0u;
  if (t == 0) out[0] = 0.0f;
}

// ---------------------------------------------------------------------------
// Kernel 1: vertical EDT scans (fg and bg simultaneously), per column.
// Writes squared column distances g_fg/g_bg; block-reduces sum(mask) per image.
// Grid: BATCH blocks x WW threads (thread = column x).
// ---------------------------------------------------------------------------
__global__ void bl_col_pass(const float* __restrict__ target,
                            float* __restrict__ g_fg,
                            float* __restrict__ g_bg,
                            float* __restrict__ imgsum) {
  const int b = blockIdx.x;
  const int x = threadIdx.x;
  const float* m = target + (size_t)b * IMG;
  float* gf = g_fg + (size_t)b * IMG;
  float* gb = g_bg + (size_t)b * IMG;

  float dfg = INF_D, dbg = INF_D, s = 0.0f;
  for (int y = 0; y < HH; ++y) {
    float mv = m[y * WW + x];
    s += mv;
    float ffg = (mv > 0.5f) ? INF_D : 0.0f;  // fg EDT seed
    float fbg = (mv > 0.5f) ? 0.0f : INF_D;  // bg EDT seed (mask of 1-m)
    dfg = fminf(ffg, dfg + 1.0f);
    dbg = fminf(fbg, dbg + 1.0f);
    gf[y * WW + x] = dfg;
    gb[y * WW + x] = dbg;
  }
  dfg = INF_D; dbg = INF_D;
  for (int y = HH - 1; y >= 0; --y) {
    float mv = m[y * WW + x];
    float ffg = (mv > 0.5f) ? INF_D : 0.0f;
    float fbg = (mv > 0.5f) ? 0.0f : INF_D;
    dfg = fminf(ffg, dfg + 1.0f);
    dbg = fminf(fbg, dbg + 1.0f);
    float vf = fminf(dfg, gf[y * WW + x]);
    float vb = fminf(dbg, gb[y * WW + x]);
    gf[y * WW + x] = vf * vf;   // squared vertical distance
    gb[y * WW + x] = vb * vb;
  }

  __shared__ float red[WW];
  red[x] = s;
  __syncthreads();
  for (int off = WW / 2; off > 0; off >>= 1) {
    if (x < off) red[x] += red[x + off];
    __syncthreads();
  }
  if (x == 0) imgsum[b] = red[0];
}

// ---------------------------------------------------------------------------
// Kernel 2: exact parabolic lower envelope along rows (min-plus, tropical).
// Row tiles are staged into LDS with CDNA5 async global->LDS copies
// (GLOBAL_LOAD_ASYNC_TO_LDS_B32, tracked by ASYNCcnt), then
// sd = sqrt(dt2_bg) - sqrt(dt2_fg); per-image max|sd| via block-max + atomicMax
// (nonnegative floats compare like their uint bit patterns).
// Grid: BATCH*HH blocks x WW threads (block = one row of one image).
// ---------------------------------------------------------------------------
__global__ void bl_row_pass(const float* __restrict__ g_fg,
                            const float* __restrict__ g_bg,
                            float* __restrict__ sd,
                            unsigned* __restrict__ maxabs) {
  const int b = blockIdx.x >> 8;
  const int y = blockIdx.x & (HH - 1);
  const int x = threadIdx.x;
  const size_t rowoff = ((size_t)b * HH + y) * WW;

  __shared__ float sf[WW];
  __shared__ float sb[WW];
  __shared__ float smax[WW];

  // Async global->LDS staging. GVS mode: mem = SGPR64 + VGPR32 + ioffset.
  // Low 32 bits of a generic pointer to __shared__ == workgroup LDS byte offset.
  {
    unsigned lds_f = (unsigned)(size_t)&sf[x];
    unsigned lds_b = (unsigned)(size_t)&sb[x];
    unsigned voff = (unsigned)((rowoff + (size_t)x) * 4u);
    asm volatile(
        "global_load_async_to_lds_b32 %0, %2, %3\n\t"
        "global_load_async_to_lds_b32 %1, %2, %4"
        :
        : "v"(lds_f), "v"(lds_b), "v"(voff), "s"(g_fg), "s"(g_bg)
        : "memory");
    asm volatile("s_wait_asynccnt 0" ::: "memory");
  }
  __syncthreads();

  float mf = 3.0e38f, mb = 3.0e38f;
  #pragma unroll 4
  for (int q = 0; q < WW; ++q) {
    float dx = (float)(x - q);
    float p = dx * dx;
    mf = fminf(mf, sf[q] + p);
    mb = fminf(mb, sb[q] + p);
  }
  float v = sqrtf(mb) - sqrtf(mf);
  sd[rowoff + x] = v;

  smax[x] = fabsf(v);
  __syncthreads();
  for (int off = WW / 2; off > 0; off >>= 1) {
    if (x < off) smax[x] = fmaxf(smax[x], smax[x + off]);
    __syncthreads();
  }
  if (x == 0) atomicMax(&maxabs[b], __float_as_uint(smax[0]));
}

// ---------------------------------------------------------------------------
// Kernel 2.5: fold validity + normalization into one per-image scale factor so
// the WMMA reduction loop has no divides/compares/divergence.
// ---------------------------------------------------------------------------
__global__ void bl_scale(const float* __restrict__ imgsum,
                         const unsigned* __restrict__ maxabs,
                         float* __restrict__ scalef) {
  int t = threadIdx.x;
  if (t < BATCH) {
    float s = imgsum[t];
    bool valid = (s > 0.0f) && (s < (float)IMG);
    float ma = __uint_as_float(maxabs[t]);
    scalef[t] = valid ? (1.0f / (ma + 1e-6f)) : 0.0f;
  }
}

// ---------------------------------------------------------------------------
// Kernel 3: final dot product sum(sigmoid(pred) * sd_normalized) / NTOT via
// V_WMMA_F32_16X16X4_F32. Each wave packs a 64-element chunk as A(16x4) of
// prob and B(4x16) of scaled sd so that diag(D) accumulates the dot product.
// Per the ISA A-layout: lane L<16 holds (K=0,K=1) of row M=L; lane L>=16
// holds (K=2,K=3) of row M=L-16 -> element indices base + 4*(L%16) + 2*(L/16)
// and +1. B uses the symmetric layout, so the per-lane addressing matches.
// 2x unroll with independent accumulators hides XDL latency; uniform control
// flow keeps EXEC all-1s (WMMA requirement).
// Grid: 64 blocks x 256 threads = 512 waves; 8192 chunk-pairs -> 16 iters/wave.
// ---------------------------------------------------------------------------
__global__ void bl_reduce_wmma(const float* __restrict__ pred,
                               const float* __restrict__ sd,
                               const float* __restrict__ scalef,
                               float* __restrict__ out) {
  const int lane = threadIdx.x & 31;
  const int wid = (int)((blockIdx.x * blockDim.x + threadIdx.x) >> 5);
  const int nwaves = (int)((gridDim.x * blockDim.x) >> 5);
  const int half = lane >> 4;
  const int m16 = lane & 15;
  const int npairs = NTOT / 128;  // two 64-element chunks per iteration

  v8f c0 = {};
  v8f c1 = {};
  for (int p = wid; p < npairs; p += nwaves) {
    int base = p * 128;
    int i0 = base + 4 * m16 + 2 * half;
    int j0 = i0 + 64;
    int b = base >> 16;  // 128 | 65536, so a pair never crosses an image
    float scale = scalef[b];

    float u0 = fast_sigmoid(pred[i0]);
    float u1 = fast_sigmoid(pred[i0 + 1]);
    float w0 = fast_sigmoid(pred[j0]);
    float w1 = fast_sigmoid(pred[j0 + 1]);

    v2f a0 = {u0, u1};
    v2f b0 = {sd[i0] * scale, sd[i0 + 1] * scale};
    v2f a1 = {w0, w1};
    v2f b1 = {sd[j0] * scale, sd[j0 + 1] * scale};

    c0 = __builtin_amdgcn_wmma_f32_16x16x4_f32(
        false, a0, false, b0, (short)0, c0, false, false);
    c1 = __builtin_amdgcn_wmma_f32_16x16x4_f32(
        false, a1, false, b1, (short)0, c1, false, false);
  }
  v8f c = c0 + c1;

  // Extract diagonal: D VGPR r holds M=r (lanes 0-15, N=lane) and M=r+8
  // (lanes 16-31, N=lane-16). (i,i): i<8 -> c[i]@lane i ; i>=8 -> c[i-8]@lane i+16.
  float carr[8];
  #pragma unroll
  for (int k = 0; k < 8; ++k) carr[k] = c[k];
  int idx = (lane < 8) ? lane : ((lane >= 24) ? (lane - 24) : -1);
  float val = (idx >= 0) ? carr[idx] : 0.0f;

  // wave32 butterfly reduce
  for (int m = 16; m > 0; m >>= 1) val += __shfl_xor(val, m, 32);
  if (lane == 0) atomicAdd(out, val * (1.0f / (float)NTOT));
}

// ---------------------------------------------------------------------------
// Host-side launcher. d_in[0]=pred f32[1048576], d_in[1]=target f32[1048576],
// d_out = 1 float. Workspace usage ~12.6 MB.
// ---------------------------------------------------------------------------
extern "C" void kernel_launch(void* const* d_in, const int* in_sizes, int n_in,
                              void* d_out, int out_size, void* d_ws, size_t ws_size,
                              hipStream_t stream) {
  const float* pred = (const float*)d_in[0];
  const float* target = (const float*)d_in[1];
  float* out = (float*)d_out;

  float* ws = (float*)d_ws;
  float* g_fg = ws;                       // NTOT floats
  float* g_bg = ws + NTOT;                // NTOT floats
  float* sd   = ws + 2 * (size_t)NTOT;    // NTOT floats
  float* imgsum = ws + 3 * (size_t)NTOT;  // BATCH floats
  unsigned* maxabs = (unsigned*)(imgsum + BATCH);  // BATCH uints
  float* scalef = (float*)(maxabs + BATCH);        // BATCH floats

  bl_init<<<1, 32, 0, stream>>>(maxabs, out);
  bl_col_pass<<<BATCH, WW, 0, stream>>>(target, g_fg, g_bg, imgsum);
  bl_row_pass<<<BATCH * HH, WW, 0, stream>>>(g_fg, g_bg, sd, maxabs);
  bl_scale<<<1, 32, 0, stream>>>(imgsum, maxabs, scalef);
  bl_reduce_wmma<<<64, 256, 0, stream>>>(pred, sd, scalef, out);
}